// FrameLogLikelihood_50629074485372
// MI455X (gfx1250) — compile-verified
//
#include <hip/hip_runtime.h>
#include <hip/hip_bf16.h>
#include <stdint.h>

// Segment-mean over static DNA frame pattern, expressed as a per-sequence GEMM
//   out[6,64] = W[6,99] * X[99,64]
// on the CDNA5 matrix pipeline with V_WMMA_F32_16X16X4_F32.
//
// One wave32 handles one (sequence, 16-column tile); 25 K-steps of 4 rows.
// The mean is folded into W (1/17 or 1/16 at pattern hits, 0 elsewhere), and
// W is a compile-time constant baked into a per-lane-laid-out table so the
// unrolled inner loop is pure {load, load, load, wmma} with immediate offsets.

typedef __attribute__((ext_vector_type(2))) float v2f;
typedef __attribute__((ext_vector_type(8))) float v8f;

#define GLOBAL_AS __attribute__((address_space(1)))

// PATTERN: kk in [0,49) -> seg kk%3 ; kk == 49 -> dropped ;
//          kk in [50,99) -> seg 3+(kk-50)%3 ; kk >= 99 -> zero pad.
// Counts: segs 0,3 have 17 rows; segs 1,2,4,5 have 16 rows.
constexpr int seg_of(int kk) {
    if (kk < 49)                  return kk % 3;
    if (kk >= 50 && kk < 99)      return 3 + (kk - 50) % 3;
    return -1;
}
constexpr float wval(int m, int kk) {
    const int s = seg_of(kk);
    if (s != m) return 0.0f;
    return (s % 3 == 0) ? (1.0f / 17.0f) : (1.0f / 16.0f);
}

// A-operand table in WMMA 16x4-f32 A layout:
//   lane 0-15  (half=0): M=lane,    VGPR0 = K=k,   VGPR1 = K=k+1
//   lane 16-31 (half=1): M=lane-16, VGPR0 = K=k+2, VGPR1 = K=k+3
// Indexed [iter][lane] -> one coalesced 256B load per wave per K-step; the
// 6.4 KB table is shared by all waves and lives in WGP$/L2.
struct alignas(16) WTab { float w[25][32][2]; };
constexpr WTab make_wtab() {
    WTab t{};
    for (int it = 0; it < 25; ++it)
        for (int lane = 0; lane < 32; ++lane) {
            const int m    = lane & 15;
            const int half = lane >> 4;
            const int kk   = it * 4 + 2 * half;
            t.w[it][lane][0] = wval(m, kk);
            t.w[it][lane][1] = wval(m, kk + 1);
        }
    return t;
}
__constant__ WTab g_wtab = make_wtab();

__global__ __launch_bounds__(128)
void segmean_wmma_kernel(const float* __restrict__ X, float* __restrict__ out) {
    const int lane  = threadIdx.x & 31;
    const int ntile = threadIdx.x >> 5;   // 0..3 : which 16-column slice of M=64
    const int seq   = blockIdx.x;         // which 99-row sequence
    const int half  = lane >> 4;          // 0: K=k,k+1   1: K=k+2,k+3
    const int mrow  = lane & 15;          // A: row M; B: column within tile
    const int n0    = ntile * 16;
    const int col   = n0 + mrow;

    const float* __restrict__ Xs = X + (size_t)seq * (99 * 64);
    // Per-lane B base: row (2*half), column col. All main-loop offsets are
    // then compile-time immediates: row it*4+2*half = base2 + it*256 floats.
    const float* __restrict__ base2 = Xs + half * 128 + col;

    // Launder the table address so the compiler computes the PC-relative base
    // exactly once into an SGPR pair (no per-iteration s_get_pc_i64 remat),
    // and pin it to the GLOBAL address space so the weight fetches lower to
    // global_load_b64 (pure LOADcnt) instead of flat_load_b64, which would
    // also increment DScnt and occupy the LDS path we never use.
    uintptr_t wp = (uintptr_t)&g_wtab;
    asm("" : "+s"(wp));
    const GLOBAL_AS v2f* __restrict__ wt = (const GLOBAL_AS v2f*)wp;

    v8f c = {};
#pragma unroll
    for (int it = 0; it < 24; ++it) {     // rows 0..95, offsets all immediate
        const v2f a = wt[it * 32 + lane];
        v2f b;
        b.x = base2[it * 256];            // row it*4 + 2*half
        b.y = base2[it * 256 + 64];       // row it*4 + 2*half + 1
        // (neg_a, A, neg_b, B, c_mod, C, reuse_a, reuse_b)
        c = __builtin_amdgcn_wmma_f32_16x16x4_f32(false, a, false, b,
                                                  (short)0, c, false, false);
    }
    {   // final K-step: rows 96..98 plus zero-weight pad row 99 clamped to 98
        const v2f a = wt[24 * 32 + lane];
        const int r0 = 96 + 2 * half;     // 96 or 98
        const int r1 = half ? 98 : 97;    // 97, or 99 clamped to 98 (w == 0)
        v2f b;
        b.x = Xs[r0 * 64 + col];
        b.y = Xs[r1 * 64 + col];
        c = __builtin_amdgcn_wmma_f32_16x16x4_f32(false, a, false, b,
                                                  (short)0, c, false, false);
    }

    // C/D 16x16 f32 layout: VGPR s, lanes 0-15 => (M=s, N=lane). Segments live
    // in rows 0..5. out[seq][s*64 + n0 + mrow].
    if (lane < 16) {
        float* __restrict__ o = out + (size_t)seq * 384 + n0 + mrow;
#pragma unroll
        for (int s = 0; s < 6; ++s)
            o[s * 64] = c[s];
    }
}

extern "C" void kernel_launch(void* const* d_in, const int* in_sizes, int n_in,
                              void* d_out, int out_size, void* d_ws, size_t ws_size,
                              hipStream_t stream) {
    const float* X = (const float*)d_in[0];
    float* out = (float*)d_out;
    const int b = in_sizes[0] / (99 * 64);   // number of sequences (8192)
    segmean_wmma_kernel<<<dim3(b), dim3(128), 0, stream>>>(X, out);
}